// TopKCELossWrapper_47519518163650
// MI455X (gfx1250) — compile-verified
//
#include <hip/hip_runtime.h>
#include <cstdint>

#define S_VOX   2097152     // 128*128*128 voxels per batch
#define TILE    1024        // voxels per staged tile
#define NTILES  4096        // 2*S_VOX / TILE
#define NBINS   1020        // histogram bins (float bits >> 21, clamped)
#define NCHUNK  255         // NBINS / 4
#define K_TOP   419430      // int(S_VOX * 0.2)
#define NBLK    1024
#define NTHR    256

// ---- CDNA5 async global->LDS DMA (gfx1250) -------------------------------
__device__ __forceinline__ void async_ld_b128(uint32_t lds_off, const void* g) {
  asm volatile("global_load_async_to_lds_b128 %0, %1, off"
               :: "v"(lds_off), "v"((unsigned long long)(uintptr_t)g)
               : "memory");
}

__device__ __forceinline__ int ce_bin(float v) {
  unsigned ub = __float_as_uint(v) >> 21;       // v >= 0: order-preserving
  return ub > (NBINS - 1) ? (NBINS - 1) : (int)ub;
}

// ---- Pass 1: fused CE + dice partials + value-weighted CE histogram -------
__global__ void __launch_bounds__(NTHR) ce_dice_pass(
    const float* __restrict__ logits, const float* __restrict__ target,
    float* __restrict__ acc, int* __restrict__ ghist, float* __restrict__ gsum)
{
  // smem[buf]: [part(2)][c(4)][TILE] floats = 32 KB per buffer.
  // Total LDS = 65536 + 2*2040*4 + 64 = 81920 B = 40 x 2KB units -> 4 blocks/WGP.
  __shared__ float smem[2][2 * 4 * TILE];
  __shared__ int   hcnt[2 * NBINS];
  __shared__ float hsum[2 * NBINS];
  __shared__ float dred[16];

  const int tid = threadIdx.x;
  for (int i = tid; i < 2 * NBINS; i += NTHR) { hcnt[i] = 0; hsum[i] = 0.0f; }
  if (tid < 16) dred[tid] = 0.0f;
  __syncthreads();

  // Stage one tile (logits + target, 4 channels each) via async DMA.
  // 2048 16B-chunks per tile -> 8 async b128 per thread (uniform, all lanes).
  // Chunk map == compute map: thread tid only ever touches its own LDS bytes,
  // so no block barrier is needed around the tile buffers.
  auto stage = [&](int t, int buf) {
    const int b  = t >> 11;             // tile -> batch (2048 tiles/batch)
    const int s0 = (t & 2047) << 10;    // voxel offset within batch
    #pragma unroll
    for (int j = 0; j < 8; ++j) {       // region j: part=j>>2, c=j&3
      const int part = j >> 2, c = j & 3;
      const float* src = (part ? target : logits)
                       + ((size_t)(b * 4 + c) * S_VOX + s0 + tid * 4);
      uint32_t loff = (uint32_t)(uintptr_t)&smem[buf][j * TILE + tid * 4];
      async_ld_b128(loff, src);
    }
  };

  stage(blockIdx.x, 0);
  int nt = 0;
  for (int t = blockIdx.x; t < NTILES; t += gridDim.x, ++nt) {
    const int  buf     = nt & 1;
    const bool hasNext = (t + (int)gridDim.x) < NTILES;   // uniform per block
    if (hasNext) {
      stage(t + gridDim.x, buf ^ 1);                      // prefetch next tile
      asm volatile("s_wait_asynccnt 0x8" ::: "memory");   // current tile done
    } else {
      asm volatile("s_wait_asynccnt 0x0" ::: "memory");
    }

    const int b = t >> 11;
    const float* Lb = &smem[buf][0];
    const float* Tb = &smem[buf][4 * TILE];

    float4 Lc[4], Tc[4];
    #pragma unroll
    for (int c = 0; c < 4; ++c) {
      Lc[c] = *(const float4*)(Lb + c * TILE + tid * 4);
      Tc[c] = *(const float4*)(Tb + c * TILE + tid * 4);
    }

    float s1l[4] = {0.f, 0.f, 0.f, 0.f};   // sum p*t per channel (this tile)
    float s2l[4] = {0.f, 0.f, 0.f, 0.f};   // sum p+t per channel
    #pragma unroll
    for (int w = 0; w < 4; ++w) {          // 4 voxels per thread
      float l[4], tg[4];
      #pragma unroll
      for (int c = 0; c < 4; ++c) {
        l[c]  = ((const float*)&Lc[c])[w];
        tg[c] = ((const float*)&Tc[c])[w];
      }
      float m = fmaxf(fmaxf(l[0], l[1]), fmaxf(l[2], l[3]));
      float e[4]; float Z = 0.f;
      #pragma unroll
      for (int c = 0; c < 4; ++c) { e[c] = __expf(l[c] - m); Z += e[c]; }
      float inv = __builtin_amdgcn_rcpf(Z);
      float ly  = l[0]*tg[0] + l[1]*tg[1] + l[2]*tg[2] + l[3]*tg[3];
      float v   = fmaxf(m + __logf(Z) - ly, 0.0f);   // CE >= 0
      const int bin = b * NBINS + ce_bin(v);
      atomicAdd(&hcnt[bin], 1);
      atomicAdd(&hsum[bin], v);
      #pragma unroll
      for (int c = 0; c < 4; ++c) {
        float p = e[c] * inv;
        s1l[c] += p * tg[c];
        s2l[c] += p + tg[c];
      }
    }
    #pragma unroll
    for (int c = 0; c < 4; ++c) {
      atomicAdd(&dred[b * 4 + c],     s1l[c]);
      atomicAdd(&dred[8 + b * 4 + c], s2l[c]);
    }
  }

  __syncthreads();   // histogram/dred are cross-wave: settle before flush
  for (int i = tid; i < 2 * NBINS; i += NTHR) {
    int v = hcnt[i];
    if (v) {
      atomicAdd(&ghist[i], v);
      atomicAdd(&gsum[i],  hsum[i]);
    }
  }
  if (tid < 16) atomicAdd(&acc[tid], dred[tid]);
}

// ---- Pass 2: chunked histogram scan + full scalar loss --------------------
__global__ void __launch_bounds__(NTHR) final_pass(
    const int* __restrict__ ghist, const float* __restrict__ gsum,
    const float* __restrict__ acc, float* __restrict__ out)
{
  __shared__ int   lcnt[NBINS];
  __shared__ float lsum[NBINS];
  __shared__ int   ccnt[NCHUNK];
  __shared__ float csum[NCHUNK];
  __shared__ float res[2];
  const int tid = threadIdx.x;

  for (int b = 0; b < 2; ++b) {
    for (int i = tid; i < NBINS; i += NTHR) {
      lcnt[i] = ghist[b * NBINS + i];
      lsum[i] = gsum[b * NBINS + i];
    }
    __syncthreads();
    if (tid < NCHUNK) {                          // 4-bin chunk totals
      int c = 0; float s = 0.f;
      #pragma unroll
      for (int j = 0; j < 4; ++j) { c += lcnt[tid * 4 + j]; s += lsum[tid * 4 + j]; }
      ccnt[tid] = c; csum[tid] = s;
    }
    __syncthreads();
    if (tid == 0) {
      int accC = 0; float accS = 0.f; int chunk = 0;
      for (int i = NCHUNK - 1; i >= 0; --i) {    // scan chunk totals, top-down
        if (accC + ccnt[i] >= K_TOP) { chunk = i; break; }
        accC += ccnt[i]; accS += csum[i];
      }
      int T = chunk * 4;
      for (int bin = chunk * 4 + 3; bin >= chunk * 4; --bin) {
        int h = lcnt[bin];
        if (accC + h >= K_TOP) { T = bin; break; }
        accC += h; accS += lsum[bin];
      }
      const int   cE = lcnt[T];
      const float sE = lsum[T];
      float need = (float)(K_TOP - accC);
      if (need < 0.f) need = 0.f;
      float est = 0.f;
      if (cE > 0) {
        // Uniform-within-bin model for the partially-selected threshold bin:
        // mean of the top f-fraction = binMean + (1-f) * width/2, clamped.
        const float lo   = __uint_as_float((unsigned)T << 21);
        const float hi   = __uint_as_float((unsigned)(T + 1) << 21);
        const float w    = hi - lo;
        const float f    = need / (float)cE;
        const float mhat = sE / (float)cE;
        est = mhat + 0.5f * (1.0f - f) * w;
        if (est > hi) est = hi;
        if (est < lo) est = lo;
      }
      res[b] = (accS + need * est) * (1.0f / (float)K_TOP);
    }
    __syncthreads();
  }

  if (tid == 0) {
    float topk = 0.5f * (res[0] + res[1]);       // mean over batches
    float dice = 0.f;
    for (int b = 0; b < 2; ++b)
      for (int c = 1; c < 4; ++c)                // skip channel 0
        dice += 1.0f - 2.0f * acc[b * 4 + c] / (acc[8 + b * 4 + c] + 1e-6f);
    dice *= (1.0f / 6.0f);
    out[0] = topk + 0.5f * dice;
  }
}

extern "C" void kernel_launch(void* const* d_in, const int* in_sizes, int n_in,
                              void* d_out, int out_size, void* d_ws, size_t ws_size,
                              hipStream_t stream)
{
  (void)in_sizes; (void)n_in; (void)out_size; (void)ws_size;
  const float* logits = (const float*)d_in[0];
  const float* target = (const float*)d_in[1];
  float* out = (float*)d_out;

  char*  ws    = (char*)d_ws;
  float* acc   = (float*)ws;                // [0..7]=s1, [8..15]=s2
  int*   ghist = (int*)(ws + 4096);         // 2*1020 counts
  float* gsum  = (float*)(ws + 16384);      // 2*1020 value sums

  hipMemsetAsync(ws, 0, 32768, stream);     // zero accumulators + histograms
  ce_dice_pass<<<NBLK, NTHR, 0, stream>>>(logits, target, acc, ghist, gsum);
  final_pass<<<1, NTHR, 0, stream>>>(ghist, gsum, acc, out);
}